// FeatureUpdateGNN_47863115546735
// MI455X (gfx1250) — compile-verified
//
#include <hip/hip_runtime.h>

// Problem constants (match the reference)
#define NNODES 50000
#define CH     128
#define NEDGES 640000
#define LSTR   132   // LDS row stride in floats: 132 % 64 == 4 -> 16 rows hit 16 distinct banks
                     // row stride in bytes = 528 = 33*16 -> every 16B chunk stays 16B-aligned

typedef __attribute__((ext_vector_type(2))) float v2f;
typedef __attribute__((ext_vector_type(8))) float v8f;

// Hardware f32 atomic add (global_atomic_add_f32), avoid CAS-loop fallback.
__device__ __forceinline__ void atom_add_f32(float* p, float v) {
  unsafeAtomicAdd(p, v);
}

// ---------------------------------------------------------------------------
// Async global->LDS staging of a 16x128 f32 tile (rows row0..row0+15) using
// CDNA5 GLOBAL_LOAD_ASYNC_TO_LDS_B128 (ASYNCcnt-tracked). Each of the 256
// threads issues 2 async 16B chunks; LDS dest uses the padded LSTR layout.
// Caller must s_wait_asynccnt 0 + __syncthreads() before reading LDS.
// ---------------------------------------------------------------------------
__device__ __forceinline__ void async_tile_load_16x128(unsigned lds_base,
                                                       const float* __restrict__ gsrc,
                                                       int row0, int t) {
  #pragma unroll
  for (int i = 0; i < 2; ++i) {
    int idx = t + i * 256;                         // 512 chunks = 16 rows * 32 x 16B
    int r   = idx >> 5;
    unsigned cb       = (unsigned)((idx & 31) << 4);               // byte col in row
    unsigned lds_addr = lds_base + (unsigned)(r * (LSTR * 4)) + cb;
    unsigned goff     = (unsigned)((row0 + r) * (CH * 4)) + cb;    // byte offset, < 2^31
    asm volatile("global_load_async_to_lds_b128 %0, %1, %2"
                 :
                 : "v"(lds_addr), "v"(goff), "s"(gsrc)
                 : "memory");
  }
}

__device__ __forceinline__ void wait_async_and_barrier() {
  asm volatile("s_wait_asynccnt 0x0" ::: "memory");
  __syncthreads();
}

// ---------------------------------------------------------------------------
// Scatter-add: one wave32 per edge, each lane handles 4 channels (float4).
// Edge indices are wave-uniform -> force into SGPRs so the 512B gather and
// the atomic scatter use saddr-form addressing. Scatter resolves as f32
// atomics in L2 (agg is 25.6MB, resident in MI455X's 192MB L2).
// addDeg: lane 0 also counts the edge into deg[dst] (first pass only).
// ---------------------------------------------------------------------------
__global__ void scatter_kernel(const float* __restrict__ feat,
                               const long long* __restrict__ src,
                               const long long* __restrict__ dst,
                               float* __restrict__ agg,
                               float* __restrict__ deg,
                               int E, int addDeg) {
  int gid  = blockIdx.x * blockDim.x + threadIdx.x;
  int e    = gid >> 5;
  int lane = gid & 31;
  if (e >= E) return;
  int s = __builtin_amdgcn_readfirstlane((int)src[e]);
  int d = __builtin_amdgcn_readfirstlane((int)dst[e]);
  const float4 v = *(const float4*)(feat + (size_t)s * CH + lane * 4);
  float* q = agg + (size_t)d * CH + lane * 4;
  atom_add_f32(q + 0, v.x);
  atom_add_f32(q + 1, v.y);
  atom_add_f32(q + 2, v.z);
  atom_add_f32(q + 3, v.w);
  if (addDeg && lane == 0) atom_add_f32(&deg[d], 1.0f);
}

// ---------------------------------------------------------------------------
// SAGE layer: out = relu( (agg/max(deg,1)) @ Wl + self @ Wr + bias )
// Block = 16 rows x 128 cols; 8 waves, each wave computes one 16x16 tile
// with v_wmma_f32_16x16x4_f32 (full f32 fidelity, 2x32 WMMAs per wave).
// Degree normalization is folded into the A fragment (scale A rows).
// Fragment layouts (ISA 7.12.2):
//   A (16x4):  M = lane&15, K = 2*(lane>>4) + vgpr
//   B (4x16):  N = lane&15, K = 2*(lane>>4) + vgpr
//   C/D:       m = vgpr + 8*(lane>>4), n = lane&15
// ---------------------------------------------------------------------------
__global__ __launch_bounds__(256)
void sage_layer_kernel(const float* __restrict__ agg,
                       const float* __restrict__ deg,
                       const float* __restrict__ self_feat,
                       const float* __restrict__ Wl,
                       const float* __restrict__ Wr,
                       const float* __restrict__ bias,
                       float* __restrict__ out) {
  __shared__ __align__(16) float sA[16 * LSTR];   // raw neighbor aggregate
  __shared__ __align__(16) float sS[16 * LSTR];   // self features
  __shared__ float sInv[16];

  const int t    = threadIdx.x;
  const int row0 = blockIdx.x * 16;

  // Async-stage both 16x128 panels straight into LDS.
  async_tile_load_16x128((unsigned)(uintptr_t)sA, agg,       row0, t);
  async_tile_load_16x128((unsigned)(uintptr_t)sS, self_feat, row0, t);
  if (t < 16) sInv[t] = 1.0f / fmaxf(deg[row0 + t], 1.0f);
  wait_async_and_barrier();

  const int lane = t & 31;
  const int wave = t >> 5;
  const int hi   = lane >> 4;       // lane-half selects K offset
  const int n15  = lane & 15;       // N for B/C frags, M for A frag
  const int col0 = wave * 16;

  const float* arow = sA + n15 * LSTR;
  const float* srow = sS + n15 * LSTR;
  const float  invm = sInv[n15];    // per-A-row mean normalization

  v8f acc = {};

  // Neighbor term: sum_k (agg[m,k]/deg[m]) * Wl[k,n]
  #pragma unroll 8
  for (int k = 0; k < CH; k += 4) {
    int ka = k + 2 * hi;
    v2f a, b;
    a.x = arow[ka]     * invm;
    a.y = arow[ka + 1] * invm;
    b.x = Wl[(size_t)ka * CH + col0 + n15];
    b.y = Wl[(size_t)(ka + 1) * CH + col0 + n15];
    acc = __builtin_amdgcn_wmma_f32_16x16x4_f32(false, a, false, b,
                                                (short)0, acc, false, false);
  }
  // Self term: sum_k self[m,k] * Wr[k,n]
  #pragma unroll 8
  for (int k = 0; k < CH; k += 4) {
    int ka = k + 2 * hi;
    v2f a, b;
    a.x = srow[ka];
    a.y = srow[ka + 1];
    b.x = Wr[(size_t)ka * CH + col0 + n15];
    b.y = Wr[(size_t)(ka + 1) * CH + col0 + n15];
    acc = __builtin_amdgcn_wmma_f32_16x16x4_f32(false, a, false, b,
                                                (short)0, acc, false, false);
  }

  const float bv = bias[col0 + n15];
  float* ocol = out + (size_t)row0 * CH + col0 + n15;
  #pragma unroll
  for (int r = 0; r < 8; ++r) {
    int m = r + 8 * hi;
    ocol[(size_t)m * CH] = fmaxf(acc[r] + bv, 0.0f);
  }
}

// ---------------------------------------------------------------------------
// Decoder: out = alpha * (h @ Wd + bd) + (1 - alpha) * x
// ---------------------------------------------------------------------------
__global__ __launch_bounds__(256)
void decoder_kernel(const float* __restrict__ h,
                    const float* __restrict__ Wd,
                    const float* __restrict__ bd,
                    const float* __restrict__ x,
                    const float* __restrict__ alpha_p,
                    float* __restrict__ out) {
  __shared__ __align__(16) float sH[16 * LSTR];

  const int t    = threadIdx.x;
  const int row0 = blockIdx.x * 16;

  async_tile_load_16x128((unsigned)(uintptr_t)sH, h, row0, t);
  wait_async_and_barrier();

  const int lane = t & 31;
  const int wave = t >> 5;
  const int hi   = lane >> 4;
  const int n15  = lane & 15;
  const int col0 = wave * 16;

  const float* arow = sH + n15 * LSTR;
  v8f acc = {};

  #pragma unroll 8
  for (int k = 0; k < CH; k += 4) {
    int ka = k + 2 * hi;
    v2f a, b;
    a.x = arow[ka];
    a.y = arow[ka + 1];
    b.x = Wd[(size_t)ka * CH + col0 + n15];
    b.y = Wd[(size_t)(ka + 1) * CH + col0 + n15];
    acc = __builtin_amdgcn_wmma_f32_16x16x4_f32(false, a, false, b,
                                                (short)0, acc, false, false);
  }

  const float al  = alpha_p[0];
  const float oma = 1.0f - al;
  const float bv  = bd[col0 + n15];
  const float* xcol = x   + (size_t)row0 * CH + col0 + n15;
  float*       ocol = out + (size_t)row0 * CH + col0 + n15;
  #pragma unroll
  for (int r = 0; r < 8; ++r) {
    int m = r + 8 * hi;
    ocol[(size_t)m * CH] = al * (acc[r] + bv) + oma * xcol[(size_t)m * CH];
  }
}

// ---------------------------------------------------------------------------
// Launcher
// ---------------------------------------------------------------------------
extern "C" void kernel_launch(void* const* d_in, const int* in_sizes, int n_in,
                              void* d_out, int out_size, void* d_ws, size_t ws_size,
                              hipStream_t stream) {
  const float*     x     = (const float*)d_in[0];
  const long long* ei    = (const long long*)d_in[1];   // int64 edge_index [2, E]
  const float*     W1l   = (const float*)d_in[2];
  const float*     b1    = (const float*)d_in[3];
  const float*     W1r   = (const float*)d_in[4];
  const float*     W2l   = (const float*)d_in[5];
  const float*     b2    = (const float*)d_in[6];
  const float*     W2r   = (const float*)d_in[7];
  const float*     Wd    = (const float*)d_in[8];
  const float*     bd    = (const float*)d_in[9];
  const float*     alpha = (const float*)d_in[10];
  float*           out   = (float*)d_out;

  const long long* src = ei;            // edge_index[0]
  const long long* dst = ei + NEDGES;   // edge_index[1]

  // Workspace layout: deg | agg (reused both layers) | h (updated in place)
  const size_t DEGF = 50176;            // NNODES rounded up; keeps agg 16B-aligned
  float* ws  = (float*)d_ws;
  float* deg = ws;
  float* agg = ws + DEGF;
  float* h   = agg + (size_t)NNODES * CH;

  const int scatter_blocks = (NEDGES * 32) / 256;   // one wave per edge
  const int gemm_blocks    = NNODES / 16;           // 50000/16 = 3125 exact

  // Zero deg + agg (contiguous)
  hipMemsetAsync(ws, 0, (DEGF + (size_t)NNODES * CH) * sizeof(float), stream);

  // Layer 1 (degree counting fused into first scatter)
  scatter_kernel<<<scatter_blocks, 256, 0, stream>>>(x, src, dst, agg, deg,
                                                     NEDGES, 1);
  sage_layer_kernel<<<gemm_blocks, 256, 0, stream>>>(agg, deg, x, W1l, W1r, b1, h);

  // Layer 2 (agg rezeroed; h updated in place — each block touches only its rows)
  hipMemsetAsync(agg, 0, (size_t)NNODES * CH * sizeof(float), stream);
  scatter_kernel<<<scatter_blocks, 256, 0, stream>>>(h, src, dst, agg, deg,
                                                     NEDGES, 0);
  sage_layer_kernel<<<gemm_blocks, 256, 0, stream>>>(agg, deg, h, W2l, W2r, b2, h);

  // Decoder + residual blend
  decoder_kernel<<<gemm_blocks, 256, 0, stream>>>(h, Wd, bd, x, alpha, out);
}